// AGSOM_50620484551281
// MI455X (gfx1250) — compile-verified
//
#include <hip/hip_runtime.h>

// AGSOM on MI455X (gfx1250): one wave32 per batch element.
// Grid state held in WMMA A-fragment layout (lane l owns node l&15,
// dims 4c+{0,1}+2*(l>=16)); per-item BMU scores computed with
// V_WMMA_F32_16X16X4_F32 chains: diag(G * (G - 2X)^T) = |g|^2 - 2 g.x.

typedef __attribute__((ext_vector_type(2))) float v2f;
typedef __attribute__((ext_vector_type(8))) float v8f;

#define SOM_LR 0.01f

__global__ __launch_bounds__(32, 1)
void agsom_wmma_kernel(const float* __restrict__ emb,
                       const float* __restrict__ nodes,
                       float* __restrict__ out,
                       int num_items)
{
    const int b    = blockIdx.x;
    const int lane = threadIdx.x;      // 0..31 (wave32)
    const int hi   = lane >> 4;        // K-half select (+2 dims)
    const int m    = lane & 15;        // flat node index owned by this lane
    const int mx   = m >> 2, my = m & 3;

    // Load grid as A fragments: A[c] = { g[m][4c+2hi], g[m][4c+2hi+1] }
    v2f A[16];
    {
        const v2f* gp = (const v2f*)(nodes + m * 64 + 2 * hi);
        #pragma unroll
        for (int c = 0; c < 16; ++c) A[c] = gp[2 * c];
    }

    const size_t bbase = (size_t)b * (size_t)num_items * 64;

    for (int pass = 0; pass < 3; ++pass) {
        for (int it = 0; it < num_items; ++it) {
            const float* xbase = emb + bbase + (size_t)it * 64;
            // warm L2/L1 ahead (speculative; OOB prefetch is dropped)
            __builtin_prefetch(xbase + 8 * 64, 0, 1);

            // x fragments: X[c] = { x[4c+2hi], x[4c+2hi+1] } (B-fragment layout,
            // columns replicated since x is the same for every N column)
            const v2f* xp = (const v2f*)(xbase + 2 * hi);
            v2f X[16];
            #pragma unroll
            for (int c = 0; c < 16; ++c) X[c] = xp[2 * c];

            // S = G * (G - 2X)^T ; diag(S)[n] = |g_n|^2 - 2 g_n.x
            // B fragment of (G-2X)^T equals (per lane) A - 2*X.
            v8f acc0 = {0.f,0.f,0.f,0.f,0.f,0.f,0.f,0.f};
            v8f acc1 = {0.f,0.f,0.f,0.f,0.f,0.f,0.f,0.f};
            #pragma unroll
            for (int c = 0; c < 16; c += 2) {
                v2f B0, B1;
                B0.x = fmaf(-2.f, X[c].x,     A[c].x);
                B0.y = fmaf(-2.f, X[c].y,     A[c].y);
                B1.x = fmaf(-2.f, X[c + 1].x, A[c + 1].x);
                B1.y = fmaf(-2.f, X[c + 1].y, A[c + 1].y);
                acc0 = __builtin_amdgcn_wmma_f32_16x16x4_f32(
                           false, A[c],     false, B0, (short)0, acc0, false, false);
                acc1 = __builtin_amdgcn_wmma_f32_16x16x4_f32(
                           false, A[c + 1], false, B1, (short)0, acc1, false, false);
            }
            v8f accs = acc0 + acc1;

            // Per-lane pick of row (lane&7)+8*hi, col m; then bpermute the
            // true diagonal element for node m into every lane.
            float a0 = (lane & 1) ? accs[1] : accs[0];
            float a1 = (lane & 1) ? accs[3] : accs[2];
            float a2 = (lane & 1) ? accs[5] : accs[4];
            float a3 = (lane & 1) ? accs[7] : accs[6];
            float b0 = (lane & 2) ? a1 : a0;
            float b1 = (lane & 2) ? a3 : a2;
            float dsel = (lane & 4) ? b1 : b0;
            int src = (m < 8) ? m : (m + 16);     // lane holding diag of node m
            float score = __shfl(dsel, src, 32);

            // Lexicographic argmin over the 16 nodes (first-index tie-break,
            // matching jnp.argmin). Halves mirror, so 4 butterfly steps suffice.
            float sc = score;
            int   bi = m;
            #pragma unroll
            for (int mask = 1; mask <= 8; mask <<= 1) {
                float os = __shfl_xor(sc, mask, 32);
                int   oi = __shfl_xor(bi, mask, 32);
                bool take = (os < sc) || ((os == sc) && (oi < bi));
                sc = take ? os : sc;
                bi = take ? oi : bi;
            }

            // 4-connected neighbor update: g += LR*(x - g) where manhattan==1
            int dx = mx - (bi >> 2); dx = dx < 0 ? -dx : dx;
            int dy = my - (bi & 3);  dy = dy < 0 ? -dy : dy;
            float f = ((dx + dy) == 1) ? SOM_LR : 0.f;
            #pragma unroll
            for (int c = 0; c < 16; ++c) {
                A[c].x = fmaf(f, X[c].x - A[c].x, A[c].x);
                A[c].y = fmaf(f, X[c].y - A[c].y, A[c].y);
            }
        }
    }

    // Sum grid over all 16 nodes: butterfly within each 16-lane group
    // (each group covers all nodes; groups own complementary dim pairs).
    #pragma unroll
    for (int c = 0; c < 16; ++c) {
        #pragma unroll
        for (int mask = 1; mask <= 8; mask <<= 1) {
            A[c].x += __shfl_xor(A[c].x, mask, 32);
            A[c].y += __shfl_xor(A[c].y, mask, 32);
        }
    }
    if (m == 0) {  // lanes 0 and 16 write their dim pairs
        v2f* op = (v2f*)(out + (size_t)b * 64 + 2 * hi);
        #pragma unroll
        for (int c = 0; c < 16; ++c) op[2 * c] = A[c];
    }
}

extern "C" void kernel_launch(void* const* d_in, const int* in_sizes, int n_in,
                              void* d_out, int out_size, void* d_ws, size_t ws_size,
                              hipStream_t stream) {
    const float* emb   = (const float*)d_in[0];   // [B, N, 64] f32
    const float* nodes = (const float*)d_in[1];   // [4, 4, 64] f32
    float* out = (float*)d_out;                   // [B, 64] f32

    const int D = 64;
    const int B = out_size / D;                   // 512
    if (B <= 0) return;
    const int num_items = in_sizes[0] / (B * D);  // 1024

    agsom_wmma_kernel<<<dim3(B), dim3(32), 0, stream>>>(emb, nodes, out, num_items);
}